// GraphInvariantPointAttention_31645319036984
// MI455X (gfx1250) — compile-verified
//
#include <hip/hip_runtime.h>
#include <hip/hip_bf16.h>
#include <math.h>

#define N_NODES 8192
#define N_EDGES 262144
#define CS_DIM  384
#define CZ_DIM  128
#define CH_DIM  16
#define H_DIM   12
#define PQK     4
#define PV      8
#define INF_VAL 100000.0f
#define EPS_VAL 1e-8f

#define S_QK 0.14433756729740643f   /* 1/sqrt(3*CH)        */
#define S_B  0.57735026918962576f   /* 1/sqrt(3)           */
#define S_PT 0.13608276348795434f   /* 1/sqrt(3*PQK*9/2)   */

typedef __attribute__((ext_vector_type(2))) float v2f;
typedef __attribute__((ext_vector_type(8))) float v8f;

// ---- monotone int encoding of float for atomicMax-based segment max ----
__device__ __forceinline__ int enc_f(float f) {
    int i = __float_as_int(f);
    return i >= 0 ? i : (i ^ 0x7fffffff);
}
__device__ __forceinline__ float dec_f(int i) {
    return __int_as_float(i >= 0 ? i : (i ^ 0x7fffffff));
}

// =====================================================================
// fp32 WMMA GEMM:  C[M,Nout] = A[M,K] @ W[K,Nout] + bias
//
// Block: 4 waves; wave w owns rows [(bx*4+w)*16, +16), all NT*16 cols of
// this block's column stripe.
//
// W is staged through LDS in 32-deep K-chunks, TRANSPOSED: WS[col][k],
// row stride 34 dwords (even -> 8B-aligned b64 reads; 34*c mod 64
// distinct for c=0..31 -> bank-conflict-free). Each B fragment is then
// a single contiguous ds_load_b64. Staging is software-pipelined: the
// next chunk's global loads are issued right after the barrier, while
// the current chunk is consumed from LDS.
//
// OOB columns: WMMA output column j depends only on B column j (lane j),
// so OOB lanes keep a clamped in-bounds garbage column and skip the
// store -> all loads unconditional, EXEC stays full.
//
// Fragment layouts (V_WMMA_F32_16X16X4_F32, wave32):
//   A 16x4 : lanes 0-15 row=l K={0,1}; lanes 16-31 row=l-16 K={2,3}
//   B 4x16 : lanes 0-15 col=l K={0,1}; lanes 16-31 col=l-16 K={2,3}
//   C 16x16: VGPR r -> (M = r+8*half, N = l15)
// Requires: K % 32 == 0, M % 64 == 0 (true for all call sites).
// =====================================================================
#define KC 32
template <int NT>
__global__ void __launch_bounds__(128)
gemm_f32_wmma(const float* __restrict__ A, const float* __restrict__ W,
              const float* __restrict__ bias, float* __restrict__ C,
              int M, int K, int Nout)
{
    __shared__ float WS[NT * 16][KC + 2];          // transposed W tile

    const int tid  = threadIdx.x;
    const int lane = tid & 31;
    const int wave = tid >> 5;
    const int m0   = (blockIdx.x * 4 + wave) * 16;
    const int n0   = blockIdx.y * (NT * 16);
    const int half = lane >> 4;
    const int l15  = lane & 15;

    v8f acc[NT];
    #pragma unroll
    for (int t = 0; t < NT; ++t)
        #pragma unroll
        for (int r = 0; r < 8; ++r) acc[t][r] = 0.0f;

    // ---- per-thread staging map: thread covers STG elements of a chunk ----
    constexpr int STG = (KC * NT * 16) / 128;
    int   rr[STG], cc[STG];
    long long woff[STG];
    #pragma unroll
    for (int i = 0; i < STG; ++i) {
        int idx = tid + i * 128;
        rr[i] = idx / (NT * 16);                   // k within chunk
        cc[i] = idx - rr[i] * (NT * 16);           // col within stripe
        int gc = n0 + cc[i];
        if (gc >= Nout) gc = Nout - 1;             // clamp: in-bounds garbage
        woff[i] = (long long)rr[i] * Nout + gc;
    }

    const float* __restrict__ arow = A + (long long)(m0 + l15) * K + 2 * half;

    // ---- prologue: fetch chunk 0 into registers ----
    float stage[STG];
    #pragma unroll
    for (int i = 0; i < STG; ++i) stage[i] = W[woff[i]];

    for (int kc = 0; kc < K; kc += KC) {
        // write staged chunk (transposed) into LDS
        #pragma unroll
        for (int i = 0; i < STG; ++i) WS[cc[i]][rr[i]] = stage[i];
        __syncthreads();

        // prefetch next chunk while computing this one
        if (kc + KC < K) {
            const float* __restrict__ wnext = W + (long long)(kc + KC) * Nout;
            #pragma unroll
            for (int i = 0; i < STG; ++i) stage[i] = wnext[woff[i]];
        }

        #pragma unroll
        for (int kk = 0; kk < KC; kk += 4) {
            v2f af;
            af.x = arow[kc + kk];
            af.y = arow[kc + kk + 1];
            #pragma unroll
            for (int t = 0; t < NT; ++t) {
                v2f bf = *(const v2f*)&WS[t * 16 + l15][kk + 2 * half];
                acc[t] = __builtin_amdgcn_wmma_f32_16x16x4_f32(
                    false, af, false, bf, (short)0, acc[t], false, false);
            }
        }
        __syncthreads();
    }

    #pragma unroll
    for (int t = 0; t < NT; ++t) {
        const int col = n0 + t * 16 + l15;
        if (col < Nout) {
            const float bs = bias[col];
            float* __restrict__ crow = C + (long long)(m0 + 8 * half) * Nout + col;
            #pragma unroll
            for (int r = 0; r < 8; ++r)
                crow[(long long)r * Nout] = acc[t][r] + bs;
        }
    }
}

// =====================================================================
// pts[n,hp,:] = rot[n] @ (y[n,0*HP+hp], y[n,1*HP+hp], y[n,2*HP+hp]) + trans[n]
// =====================================================================
__global__ void to_pts_kernel(const float* __restrict__ y,
                              const float* __restrict__ rot,
                              const float* __restrict__ trans,
                              float* __restrict__ pts, int HP, int total)
{
    int idx = blockIdx.x * blockDim.x + threadIdx.x;
    if (idx >= total) return;
    int n = idx / HP, hp = idx - n * HP;
    const float* __restrict__ yr = y + (long long)n * (3 * HP);
    float x  = yr[hp];
    float yy = yr[HP + hp];
    float zz = yr[2 * HP + hp];
    const float* __restrict__ R = rot + n * 9;
    const float* __restrict__ t = trans + n * 3;
    float* __restrict__ o = pts + (long long)idx * 3;
    o[0] = R[0] * x + R[1] * yy + R[2] * zz + t[0];
    o[1] = R[3] * x + R[4] * yy + R[5] * zz + t[1];
    o[2] = R[6] * x + R[7] * yy + R[8] * zz + t[2];
}

// =====================================================================
// Zero accumulators (denom|o|o_pt|o_pair contiguous) and seed amax=enc(-inf).
// =====================================================================
__global__ void init_kernel(float* __restrict__ acc, int zn,
                            int* __restrict__ amax, int an)
{
    int i = blockIdx.x * blockDim.x + threadIdx.x;
    if (i < zn) acc[i] = 0.0f;
    if (i < an) amax[i] = (int)0x807FFFFF;   // enc(-inf)
}

// =====================================================================
// Edge logits + atomic running max per (src, h).
// =====================================================================
__global__ void edge_logits_kernel(const int* __restrict__ ei,
    const float* __restrict__ q, const float* __restrict__ kv,
    const float* __restrict__ q_pts, const float* __restrict__ kv_pts,
    const float* __restrict__ bmat, const float* __restrict__ mask,
    const float* __restrict__ head_w,
    float* __restrict__ alog, int* __restrict__ amax)
{
    int t = blockIdx.x * blockDim.x + threadIdx.x;
    if (t >= N_EDGES * H_DIM) return;
    int e = t / H_DIM, h = t - e * H_DIM;
    int dst = ei[e];
    int src = ei[N_EDGES + e];

    const float4* __restrict__ qr = (const float4*)(q  + ((long long)src * H_DIM + h) * CH_DIM);
    const float4* __restrict__ kr = (const float4*)(kv + ((long long)dst * H_DIM + h) * (2 * CH_DIM));
    float dot = 0.0f;
    #pragma unroll
    for (int c = 0; c < 4; ++c) {
        float4 a4 = qr[c], b4 = kr[c];
        dot += a4.x * b4.x + a4.y * b4.y + a4.z * b4.z + a4.w * b4.w;
    }

    const float* __restrict__ qp = q_pts  + ((long long)src * H_DIM + h) * (PQK * 3);
    const float* __restrict__ kp = kv_pts + ((long long)dst * H_DIM + h) * ((PQK + PV) * 3);
    float d2 = 0.0f;
    #pragma unroll
    for (int j = 0; j < PQK * 3; ++j) { float df = qp[j] - kp[j]; d2 += df * df; }

    float hw = logf(1.0f + expf(head_w[h])) * S_PT;
    float a  = dot * S_QK
             + S_B * bmat[(long long)e * H_DIM + h]
             - 0.5f * hw * d2
             + INF_VAL * (mask[dst] * mask[src] - 1.0f);

    alog[t] = a;
    atomicMax(&amax[src * H_DIM + h], enc_f(a));
}

__global__ void amax_fix_kernel(const int* __restrict__ amax,
                                float* __restrict__ amaxf, int n)
{
    int i = blockIdx.x * blockDim.x + threadIdx.x;
    if (i >= n) return;
    float f = dec_f(amax[i]);
    amaxf[i] = isfinite(f) ? f : 0.0f;
}

// =====================================================================
// ex = exp(a - amax[src]); scatter ex and ex-weighted v / v_pts / pair_z.
// Normalization by sum(ex) is deferred to the node pass (linear op).
// =====================================================================
__global__ void edge_accum_kernel(const int* __restrict__ ei,
    const float* __restrict__ alog, const float* __restrict__ amaxf,
    const float* __restrict__ kv, const float* __restrict__ kv_pts,
    const float* __restrict__ pair_z,
    float* __restrict__ denom, float* __restrict__ o,
    float* __restrict__ o_pt, float* __restrict__ o_pair)
{
    int t = blockIdx.x * blockDim.x + threadIdx.x;
    if (t >= N_EDGES * H_DIM) return;
    int e = t / H_DIM, h = t - e * H_DIM;
    int dst = ei[e];
    int src = ei[N_EDGES + e];

    float ex = expf(alog[t] - amaxf[src * H_DIM + h]);
    atomicAdd(&denom[src * H_DIM + h], ex);

    const float* __restrict__ vr = kv + ((long long)dst * H_DIM + h) * (2 * CH_DIM) + CH_DIM;
    float* __restrict__ orow = o + ((long long)src * H_DIM + h) * CH_DIM;
    #pragma unroll
    for (int c = 0; c < CH_DIM; ++c) atomicAdd(&orow[c], ex * vr[c]);

    const float* __restrict__ vp = kv_pts + ((long long)dst * H_DIM + h) * ((PQK + PV) * 3) + PQK * 3;
    float* __restrict__ optr = o_pt + ((long long)src * H_DIM + h) * (PV * 3);
    #pragma unroll
    for (int j = 0; j < PV * 3; ++j) atomicAdd(&optr[j], ex * vp[j]);

    const float* __restrict__ pz = pair_z + (long long)e * 32;
    float* __restrict__ opr = o_pair + ((long long)src * H_DIM + h) * 32;
    #pragma unroll
    for (int j = 0; j < 32; ++j) atomicAdd(&opr[j], ex * pz[j]);
}

// =====================================================================
// Per (n,h): scale by 1/denom, inverse frame transform, norms, pack feats.
// feats layout: [o(192) | pt_x(96) | pt_y(96) | pt_z(96) | pt_norm(96) | o_pair(384)]
// =====================================================================
__global__ void node_finalize_kernel(const float* __restrict__ denom,
    const float* __restrict__ o, const float* __restrict__ o_pt,
    const float* __restrict__ o_pair, const float* __restrict__ rot,
    const float* __restrict__ trans, float* __restrict__ feats)
{
    int t = blockIdx.x * blockDim.x + threadIdx.x;
    if (t >= N_NODES * H_DIM) return;
    int n = t / H_DIM, h = t - n * H_DIM;
    float scale = 1.0f / (denom[t] + 1e-16f);
    float* __restrict__ f = feats + (long long)n * 960;

    const float* __restrict__ orow = o + (long long)t * CH_DIM;
    #pragma unroll
    for (int c = 0; c < CH_DIM; ++c) f[h * CH_DIM + c] = orow[c] * scale;

    const float* __restrict__ R  = rot + n * 9;
    const float* __restrict__ tr = trans + n * 3;
    const float* __restrict__ op = o_pt + (long long)t * (PV * 3);
    #pragma unroll
    for (int p = 0; p < PV; ++p) {
        float gx = op[p * 3 + 0] * scale - tr[0];
        float gy = op[p * 3 + 1] * scale - tr[1];
        float gz = op[p * 3 + 2] * scale - tr[2];
        // out_i = sum_j R[j][i] * g_j  (R^T)
        float lx = R[0] * gx + R[3] * gy + R[6] * gz;
        float ly = R[1] * gx + R[4] * gy + R[7] * gz;
        float lz = R[2] * gx + R[5] * gy + R[8] * gz;
        int hp = h * PV + p;
        f[192 + hp]       = lx;
        f[192 + 96 + hp]  = ly;
        f[192 + 192 + hp] = lz;
        f[192 + 288 + hp] = sqrtf(lx * lx + ly * ly + lz * lz + EPS_VAL);
    }

    const float* __restrict__ pr = o_pair + (long long)t * 32;
    #pragma unroll
    for (int j = 0; j < 32; ++j) f[576 + h * 32 + j] = pr[j] * scale;
}

// =====================================================================
extern "C" void kernel_launch(void* const* d_in, const int* in_sizes, int n_in,
                              void* d_out, int out_size, void* d_ws, size_t ws_size,
                              hipStream_t stream)
{
    (void)in_sizes; (void)n_in; (void)out_size; (void)ws_size;

    const float* s     = (const float*)d_in[0];
    const float* z     = (const float*)d_in[1];
    const int*   ei    = (const int*)  d_in[2];
    const float* rot   = (const float*)d_in[3];
    const float* trans = (const float*)d_in[4];
    const float* mask  = (const float*)d_in[5];
    const float* q_w   = (const float*)d_in[6];
    const float* q_b   = (const float*)d_in[7];
    const float* kv_w  = (const float*)d_in[8];
    const float* kv_b  = (const float*)d_in[9];
    const float* qp_w  = (const float*)d_in[10];
    const float* qp_b  = (const float*)d_in[11];
    const float* kvp_w = (const float*)d_in[12];
    const float* kvp_b = (const float*)d_in[13];
    const float* b_w   = (const float*)d_in[14];
    const float* b_b   = (const float*)d_in[15];
    const float* dz_w  = (const float*)d_in[16];
    const float* dz_b  = (const float*)d_in[17];
    const float* hwts  = (const float*)d_in[18];
    const float* out_w = (const float*)d_in[19];
    const float* out_b = (const float*)d_in[20];
    float* out = (float*)d_out;

    // ---- workspace carve-up (fp32 elements, 256B aligned chunks) ----
    float* ws = (float*)d_ws;
    size_t off = 0;
    auto carve = [&](size_t nelem) {
        float* p = ws + off;
        off += (nelem + 63) & ~(size_t)63;
        return p;
    };
    const size_t NH = (size_t)N_NODES * H_DIM;
    float* q      = carve((size_t)N_NODES * H_DIM * CH_DIM);            // 1.57M
    float* kv     = carve((size_t)N_NODES * H_DIM * 2 * CH_DIM);        // 3.15M
    float* q_pts  = carve((size_t)N_NODES * H_DIM * PQK * 3);           // 1.18M
    float* kv_pts = carve((size_t)N_NODES * H_DIM * (PQK + PV) * 3);    // 3.54M
    float* bmat   = carve((size_t)N_EDGES * H_DIM);                     // 3.15M
    float* pair_z = carve((size_t)N_EDGES * 32);                        // 8.39M
    float* alog   = carve((size_t)N_EDGES * H_DIM);                     // 3.15M
    int*   amax   = (int*)carve(NH);
    float* amaxf  = carve(NH);
    float* accum  = carve(NH * 73);              // denom | o | o_pt | o_pair
    float* denom  = accum;
    float* o      = denom + NH;
    float* o_pt   = o + NH * CH_DIM;
    float* o_pair = o_pt + NH * PV * 3;
    float* feats  = carve((size_t)N_NODES * 960);                       // 7.86M
    // temp GEMM outputs for point projections alias the (not-yet-written) feats
    float* y_qp   = feats;
    float* y_kvp  = feats + (size_t)N_NODES * H_DIM * PQK * 3;

    dim3 gblk(128);
    auto gemm = [&](const float* A, const float* W, const float* bias, float* C,
                    int M, int K, int Nout) {
        if (Nout >= 64) {
            dim3 grid(M / 64, (Nout + 63) / 64);
            hipLaunchKernelGGL((gemm_f32_wmma<4>), grid, gblk, 0, stream, A, W, bias, C, M, K, Nout);
        } else {
            dim3 grid(M / 64, (Nout + 15) / 16);
            hipLaunchKernelGGL((gemm_f32_wmma<1>), grid, gblk, 0, stream, A, W, bias, C, M, K, Nout);
        }
    };

    // 1) node projections (fp32 WMMA)
    gemm(s, q_w,   q_b,   q,     N_NODES, CS_DIM, H_DIM * CH_DIM);
    gemm(s, kv_w,  kv_b,  kv,    N_NODES, CS_DIM, 2 * H_DIM * CH_DIM);
    gemm(s, qp_w,  qp_b,  y_qp,  N_NODES, CS_DIM, H_DIM * PQK * 3);
    gemm(s, kvp_w, kvp_b, y_kvp, N_NODES, CS_DIM, H_DIM * (PQK + PV) * 3);

    // 2) edge projections (fp32 WMMA)
    gemm(z, b_w,  b_b,  bmat,   N_EDGES, CZ_DIM, H_DIM);
    gemm(z, dz_w, dz_b, pair_z, N_EDGES, CZ_DIM, 32);

    // 3) frame transforms
    {
        int tq = N_NODES * H_DIM * PQK;
        int tk = N_NODES * H_DIM * (PQK + PV);
        to_pts_kernel<<<(tq + 255) / 256, 256, 0, stream>>>(y_qp,  rot, trans, q_pts,  H_DIM * PQK,        tq);
        to_pts_kernel<<<(tk + 255) / 256, 256, 0, stream>>>(y_kvp, rot, trans, kv_pts, H_DIM * (PQK + PV), tk);
    }

    // 4) re-init accumulators every call (graph replays must be idempotent)
    {
        int zn = (int)(NH * 73);
        init_kernel<<<(zn + 255) / 256, 256, 0, stream>>>(accum, zn, amax, (int)NH);
    }

    // 5) attention logits + segment max (atomicMax on ordered-int floats)
    const int EH = N_EDGES * H_DIM;
    edge_logits_kernel<<<(EH + 255) / 256, 256, 0, stream>>>(
        ei, q, kv, q_pts, kv_pts, bmat, mask, hwts, alog, amax);
    amax_fix_kernel<<<((int)NH + 255) / 256, 256, 0, stream>>>(amax, amaxf, (int)NH);

    // 6) exp + unnormalized scatter-aggregation (L2 fp32 atomics)
    edge_accum_kernel<<<(EH + 255) / 256, 256, 0, stream>>>(
        ei, alog, amaxf, kv, kv_pts, pair_z, denom, o, o_pt, o_pair);

    // 7) per-node finalize -> feats
    node_finalize_kernel<<<((int)NH + 255) / 256, 256, 0, stream>>>(
        denom, o, o_pt, o_pair, rot, trans, feats);

    // 8) output projection (fp32 WMMA): out = feats @ out_w + out_b
    gemm(feats, out_w, out_b, out, N_NODES, 960, CS_DIM);
}